// MolecularGraphNeuralNetwork_360777253657
// MI455X (gfx1250) — compile-verified
//
#include <hip/hip_runtime.h>
#include <hip/hip_bf16.h>
#include <math.h>

typedef __attribute__((ext_vector_type(2))) float    v2f;
typedef __attribute__((ext_vector_type(8))) float    v8f;
typedef __attribute__((ext_vector_type(4))) unsigned v4u;
typedef __attribute__((ext_vector_type(8))) int      v8i;
typedef __attribute__((ext_vector_type(4))) int      v4i;

#define DIMD 128
#define NPM  64
#define NTOT 16384

__device__ __forceinline__ float gelu_tanh(float x) {
    const float c = 0.7978845608028654f; // sqrt(2/pi)
    float u = c * (x + 0.044715f * x * x * x);
    // 0.5*x*(1+tanh(u)) == x * sigmoid(2u) == x / (1 + exp(-2u))
    return x / (1.0f + __expf(-2.0f * u));
}

// per-lane async global -> LDS, 16 bytes (ASYNCcnt)
__device__ __forceinline__ void async_g2l_b128(void* lds_ptr, const void* gptr) {
    unsigned           lds = (unsigned)(uintptr_t)lds_ptr;          // low 32 bits = LDS offset
    unsigned long long ga  = (unsigned long long)(uintptr_t)gptr;
    asm volatile("global_load_async_to_lds_b128 %0, %1, off"
                 :: "v"(lds), "v"(ga) : "memory");
}
__device__ __forceinline__ void wait_async0() {
    asm volatile("s_wait_asynccnt 0" ::: "memory");
}

// TDM: DMA a 128x128 fp32 row-major tile (row stride 128) from `gaddr` into LDS
__device__ __forceinline__ void tdm_load_w_tile(void* lds_ptr, const void* gaddr) {
    unsigned long long ga = (unsigned long long)(uintptr_t)gaddr;
    v4u g0;
    g0.x = 1u;                                            // count=1, user descriptor
    g0.y = (unsigned)(uintptr_t)lds_ptr;                  // lds_addr
    g0.z = (unsigned)(ga & 0xFFFFFFFFull);                // global_addr[31:0]
    g0.w = (unsigned)((ga >> 32) & 0x1FFFFFFull) | (2u << 30); // global_addr[56:32] | type=2
    v8i g1;
    g1[0] = (int)(2u << 16);       // data_size=2 (4B), wg_mask=0
    g1[1] = (int)(128u << 16);     // tensor_dim0 = 128 (low 16 in [63:48])
    g1[2] = (int)(128u << 16);     // tensor_dim0 hi=0, tensor_dim1=128 (low 16 in [95:80])
    g1[3] = (int)(128u << 16);     // tensor_dim1 hi=0, tile_dim0=128
    g1[4] = 128;                   // tile_dim1=128, tile_dim2=0
    g1[5] = 128;                   // tensor_dim0_stride[31:0] = 128
    g1[6] = 0;                     // dim0_stride hi, dim1_stride lo
    g1[7] = 0;                     // dim1_stride hi
    v4i z4 = {0, 0, 0, 0};         // groups 2/3: dims beyond 2D unused (tile_dim3=0)
    v8i z8 = {0, 0, 0, 0, 0, 0, 0, 0};
    __builtin_amdgcn_tensor_load_to_lds(g0, g1, z4, z4, z8, 0);
}

__global__ __launch_bounds__(256) void
mgnn_mol_kernel(const int*   __restrict__ fingerprints, // [16384]
                const float* __restrict__ adjacency,    // [16384,16384] block-diag
                const float* __restrict__ embed,        // [NF,128]
                const float* __restrict__ Wfp,          // [3,128,128]
                const float* __restrict__ bfp,          // [3,128]
                const float* __restrict__ Wout,         // [2,128,128]
                const float* __restrict__ bout,         // [2,128]
                const float* __restrict__ masks,        // [256,1]
                float*       __restrict__ out)          // [256,128]
{
    __shared__ float sV[NPM * DIMD];    // 32 KB  current node features
    __shared__ float sH[NPM * DIMD];    // 32 KB  gelu(V W^T + b)
    __shared__ float sA[NPM * NPM];     // 16 KB  adjacency diagonal block
    __shared__ float sW[DIMD * DIMD];   // 64 KB  current layer weight (TDM target)
    __shared__ float sMol[2 * DIMD];    // ping-pong molecule vector

    const int mol  = blockIdx.x;
    const int t    = threadIdx.x;
    const int lane = t & 31;
    const int wave = t >> 5;

    // ---- async-stage embeddings: 4 threads/row, 8x b128 per thread (gathered rows)
    {
        int row  = t >> 2;
        int part = t & 3;
        int fp   = fingerprints[mol * NPM + row];
        const float* src = embed + (size_t)fp * DIMD + part * 32;
        float*       dst = sV + row * DIMD + part * 32;
#pragma unroll
        for (int i = 0; i < 8; ++i) async_g2l_b128(dst + i * 4, src + i * 4);
    }
    // ---- async-stage 64x64 adjacency diagonal block (rest of matrix is exactly 0)
    {
        int row  = t >> 2;
        int part = t & 3;
        const float* src = adjacency + (size_t)(mol * NPM + row) * NTOT
                                     + (size_t)mol * NPM + part * 16;
        float*       dst = sA + row * NPM + part * 16;
#pragma unroll
        for (int i = 0; i < 4; ++i) async_g2l_b128(dst + i * 4, src + i * 4);
    }
    __builtin_prefetch(Wout, 0, 0);   // warm GL2 for the tail MLP weights
    wait_async0();

    const int mt   = wave & 3;          // M-tile 0..3 (rows 16*mt..)
    const int ntb  = (wave >> 2) * 4;   // first of 4 N-tiles
    const int cl   = lane & 15;         // column/row within tile half
    const int hi   = lane >> 4;         // 0: lanes 0-15, 1: lanes 16-31
    const int koff = hi * 2;            // K offset of this lane's fragment pair
    const int arow = mt * 16 + cl;      // A-fragment row (M)

    for (int l = 0; l < 3; ++l) {
        // TDM-stage this layer's 128x128 weight (row-major W[n][k]) into sW
        if (wave == 0) {
            tdm_load_w_tile(sW, Wfp + (size_t)l * DIMD * DIMD);
            __builtin_amdgcn_s_wait_tensorcnt(0);
        }
        __syncthreads();

        // ===== GEMM1: H = gelu(V @ W^T + b)   M=64 N=128 K=128, f32 WMMA =====
        {
            v8f acc[4];
#pragma unroll
            for (int j = 0; j < 4; ++j) acc[j] = (v8f)0.0f;

            for (int k = 0; k < DIMD; k += 4) {
                v2f a = *(const v2f*)(sV + arow * DIMD + k + koff);
#pragma unroll
                for (int j = 0; j < 4; ++j) {
                    int n = (ntb + j) * 16 + cl;
                    v2f b = *(const v2f*)(sW + n * DIMD + k + koff); // B[kb][n] = W[n][kb]
                    acc[j] = __builtin_amdgcn_wmma_f32_16x16x4_f32(
                        false, a, false, b, (short)0, acc[j], false, false);
                }
            }
#pragma unroll
            for (int j = 0; j < 4; ++j) {
#pragma unroll
                for (int r = 0; r < 8; ++r) {
                    int row = mt * 16 + r + (hi << 3);
                    int col = (ntb + j) * 16 + cl;
                    float x = acc[j][r] + bfp[l * DIMD + col];
                    sH[row * DIMD + col] = gelu_tanh(x);
                }
            }
        }
        __syncthreads();

        // ===== GEMM2: V = H + A @ H   M=64 N=128 K=64 =====
        {
            v8f acc[4];
#pragma unroll
            for (int j = 0; j < 4; ++j) {      // residual init: C = H tile
#pragma unroll
                for (int r = 0; r < 8; ++r) {
                    int row = mt * 16 + r + (hi << 3);
                    int col = (ntb + j) * 16 + cl;
                    acc[j][r] = sH[row * DIMD + col];
                }
            }
            for (int k = 0; k < NPM; k += 4) {
                v2f a = *(const v2f*)(sA + arow * NPM + k + koff);
#pragma unroll
                for (int j = 0; j < 4; ++j) {
                    int n  = (ntb + j) * 16 + cl;
                    int kb = k + koff;
                    v2f b;
                    b.x = sH[(kb + 0) * DIMD + n];
                    b.y = sH[(kb + 1) * DIMD + n];
                    acc[j] = __builtin_amdgcn_wmma_f32_16x16x4_f32(
                        false, a, false, b, (short)0, acc[j], false, false);
                }
            }
#pragma unroll
            for (int j = 0; j < 4; ++j) {
#pragma unroll
                for (int r = 0; r < 8; ++r) {
                    int row = mt * 16 + r + (hi << 3);
                    int col = (ntb + j) * 16 + cl;
                    sV[row * DIMD + col] = acc[j][r];
                }
            }
        }
        __syncthreads();

        // ===== row-wise L2 normalize (thread r owns row r) =====
        if (t < NPM) {
            float s = 0.0f;
#pragma unroll 4
            for (int c = 0; c < DIMD; ++c) { float v = sV[t * DIMD + c]; s += v * v; }
            float inv = 1.0f / fmaxf(sqrtf(s), 1e-12f);
#pragma unroll 4
            for (int c = 0; c < DIMD; ++c) sV[t * DIMD + c] *= inv;
        }
        __syncthreads();
    }

    // ===== segment-sum pool over the molecule's 64 nodes =====
    if (t < DIMD) {
        float s = 0.0f;
#pragma unroll 4
        for (int r = 0; r < NPM; ++r) s += sV[r * DIMD + t];
        sMol[t] = s;
    }
    __syncthreads();

    // ===== two output layers: mol = gelu(mol @ Wout^T + b) =====
    int cur = 0;
    for (int l = 0; l < 2; ++l) {
        if (t < DIMD) {
            const float* w = Wout + ((size_t)l * DIMD + t) * DIMD;
            float s = bout[l * DIMD + t];
#pragma unroll 4
            for (int k = 0; k < DIMD; ++k) s += sMol[cur * DIMD + k] * w[k];
            sMol[(1 - cur) * DIMD + t] = gelu_tanh(s);
        }
        __syncthreads();
        cur ^= 1;
    }

    if (t < DIMD) out[(size_t)mol * DIMD + t] = sMol[cur * DIMD + t] * masks[mol];
}

extern "C" void kernel_launch(void* const* d_in, const int* in_sizes, int n_in,
                              void* d_out, int out_size, void* d_ws, size_t ws_size,
                              hipStream_t stream) {
    (void)n_in; (void)d_ws; (void)ws_size; (void)out_size;
    const int*   fingerprints = (const int*)  d_in[0];
    const float* adjacency    = (const float*)d_in[1];
    const float* embed        = (const float*)d_in[2];
    const float* Wfp          = (const float*)d_in[3];
    const float* bfp          = (const float*)d_in[4];
    const float* Wout         = (const float*)d_in[5];
    const float* bout         = (const float*)d_in[6];
    const float* masks        = (const float*)d_in[7];
    float*       out          = (float*)d_out;

    const int Bmol = in_sizes[7]; // masks: [B,1] -> 256 molecules
    dim3 grid(Bmol), block(256);
    mgnn_mol_kernel<<<grid, block, 0, stream>>>(fingerprints, adjacency, embed,
                                                Wfp, bfp, Wout, bout, masks, out);
}